// GatedGCNLayer_69269232550020
// MI455X (gfx1250) — compile-verified
//
#include <hip/hip_runtime.h>
#include <hip/hip_bf16.h>

// GatedGCN layer for MI455X (gfx1250, wave32, WMMA, async global->LDS).
// D = 128 fixed. fp32 end-to-end using V_WMMA_F32_16X16X4_F32.

#define D 128
#define KSTEPS (D / 4)   // 32 wmma k-steps per 128-deep dot product

typedef __attribute__((ext_vector_type(2))) float v2f;
typedef __attribute__((ext_vector_type(8))) float v8f;

// ---------------------------------------------------------------------------
// gfx1250 async copy: global memory -> LDS, 16 bytes per lane, ASYNCcnt-
// tracked, bypasses VGPRs (CDNA5 ISA, GV addressing mode).
// lds_off is a byte offset from the wave's LDS base (low 32 bits of a
// generic pointer to __shared__).
// ---------------------------------------------------------------------------
__device__ __forceinline__ void async_copy_b128(unsigned lds_off,
                                                const float* gaddr)
{
    unsigned long long ga = (unsigned long long)gaddr;
    asm volatile("global_load_async_to_lds_b128 %0, %1, off"
                 :: "v"(lds_off), "v"(ga)
                 : "memory");
}

__device__ __forceinline__ void wait_async0()
{
    asm volatile("s_wait_asynccnt 0" ::: "memory");
}

__device__ __forceinline__ unsigned lds_offset_of(const void* p)
{
    // Generic shared pointers on AMDGPU are {SHARED_BASE, 32-bit LDS offset}.
    return (unsigned)(unsigned long long)p;
}

// ---------------------------------------------------------------------------
// Node projection: out[n,:] = h[n,:] @ W + b      (N rows, W is DxD row-major)
// Block = 256 threads = 8 waves; wave w owns output columns [16w, 16w+16).
// B slice (128x16) resident in 64 VGPRs; A-tile double-buffered in LDS via
// async copies overlapping the WMMA chain of the current tile.
// ---------------------------------------------------------------------------
__global__ __launch_bounds__(256)
void node_proj_kernel(const float* __restrict__ h,
                      const float* __restrict__ W,
                      const float* __restrict__ b,
                      float* __restrict__ out,
                      int nrows)
{
    __shared__ float sh[2][16 * D];

    const int lane = threadIdx.x & 31;
    const int wave = threadIdx.x >> 5;      // 0..7
    const int half = lane >> 4;             // 0 or 1
    const int n0   = wave * 16;
    const int col  = n0 + (lane & 15);      // this lane's output column

    // Resident B fragments: bfrag[k] = { W[4k+2*half][col], W[4k+2*half+1][col] }
    v2f bfrag[KSTEPS];
#pragma unroll
    for (int k = 0; k < KSTEPS; ++k) {
        const int kb = 4 * k + 2 * half;
        bfrag[k].x = W[(size_t)kb * D + col];
        bfrag[k].y = W[(size_t)(kb + 1) * D + col];
    }
    const float bias = b[col];

    // Per-thread copy slice: 32 contiguous bytes (2 x b128) of the 8 KB tile.
    const unsigned my_elt  = threadIdx.x * 8;            // float index in tile
    const unsigned lds0    = lds_offset_of(&sh[0][my_elt]);
    const unsigned lds1    = lds_offset_of(&sh[1][my_elt]);

    const int ntiles = nrows >> 4;          // 16-row tiles (N % 16 == 0)
    const int stride = gridDim.x;
    int t = blockIdx.x;
    if (t >= ntiles) return;

    // Prologue: async-stage the first tile into buffer 0.
    {
        const float* gp = h + (size_t)t * 16 * D + my_elt;
        async_copy_b128(lds0, gp);
        async_copy_b128(lds0 + 16, gp + 4);
    }
    wait_async0();
    __syncthreads();

    int buf = 0;
    for (; t < ntiles; t += stride) {
        // Kick off the async copy of the next tile into the spare buffer.
        const int tn = t + stride;
        if (tn < ntiles) {
            const float* gp = h + (size_t)tn * 16 * D + my_elt;
            const unsigned l = (buf ? lds0 : lds1);
            async_copy_b128(l, gp);
            async_copy_b128(l + 16, gp + 4);
        }

        const float* shc = sh[buf];

        v8f acc;
#pragma unroll
        for (int r = 0; r < 8; ++r) acc[r] = bias;

#pragma unroll
        for (int k = 0; k < KSTEPS; ++k) {
            const v2f a = *(const v2f*)&shc[(lane & 15) * D + 4 * k + 2 * half];
            acc = __builtin_amdgcn_wmma_f32_16x16x4_f32(
                false, a, false, bfrag[k], (short)0, acc, false, false);
        }

        // C/D layout: VGPR r holds row (r + 8*half), column (lane & 15) + n0.
#pragma unroll
        for (int r = 0; r < 8; ++r) {
            const int row = t * 16 + r + 8 * half;
            out[(size_t)row * D + col] = acc[r];
        }

        wait_async0();        // our async writes into the spare buffer landed
        __syncthreads();      // everyone done reading buf / writing spare
        buf ^= 1;
    }
}

// ---------------------------------------------------------------------------
// Fused edge kernel: Ce = e@WC + bC (WMMA), then per element
//   e_ij  = Ce + Dh[src] + Eh[dst]
//   sigma = sigmoid(e_ij)
//   e_out = e + relu(e_ij)
//   num[dst] += sigma * Bh[src]   (global fp32 atomics, L2-resident tables)
//   den[dst] += sigma
// Reads e exactly once (async-staged LDS tile reused for the residual add).
// ---------------------------------------------------------------------------
__global__ __launch_bounds__(256)
void edge_kernel(const float* __restrict__ e,
                 const int*   __restrict__ src,
                 const int*   __restrict__ dst,
                 const float* __restrict__ WC,
                 const float* __restrict__ bC,
                 const float* __restrict__ Bh,
                 const float* __restrict__ Dh,
                 const float* __restrict__ Eh,
                 float* __restrict__ num,
                 float* __restrict__ den,
                 float* __restrict__ e_out,
                 int nedges)
{
    __shared__ float sh[2][16 * D];

    const int lane = threadIdx.x & 31;
    const int wave = threadIdx.x >> 5;
    const int half = lane >> 4;
    const int n0   = wave * 16;
    const int col  = n0 + (lane & 15);

    v2f bfrag[KSTEPS];
#pragma unroll
    for (int k = 0; k < KSTEPS; ++k) {
        const int kb = 4 * k + 2 * half;
        bfrag[k].x = WC[(size_t)kb * D + col];
        bfrag[k].y = WC[(size_t)(kb + 1) * D + col];
    }
    const float bias = bC[col];

    const unsigned my_elt = threadIdx.x * 8;
    const unsigned lds0   = lds_offset_of(&sh[0][my_elt]);
    const unsigned lds1   = lds_offset_of(&sh[1][my_elt]);

    const int ntiles = nedges >> 4;         // E % 16 == 0
    const int stride = gridDim.x;
    int t = blockIdx.x;
    if (t >= ntiles) return;

    {
        const float* gp = e + (size_t)t * 16 * D + my_elt;
        async_copy_b128(lds0, gp);
        async_copy_b128(lds0 + 16, gp + 4);
    }
    wait_async0();
    __syncthreads();

    int buf = 0;
    for (; t < ntiles; t += stride) {
        const int tn = t + stride;
        if (tn < ntiles) {
            const float* gp = e + (size_t)tn * 16 * D + my_elt;
            const unsigned l = (buf ? lds0 : lds1);
            async_copy_b128(l, gp);
            async_copy_b128(l + 16, gp + 4);
        }

        const float* shc = sh[buf];

        v8f acc;
#pragma unroll
        for (int r = 0; r < 8; ++r) acc[r] = bias;

#pragma unroll
        for (int k = 0; k < KSTEPS; ++k) {
            const v2f a = *(const v2f*)&shc[(lane & 15) * D + 4 * k + 2 * half];
            acc = __builtin_amdgcn_wmma_f32_16x16x4_f32(
                false, a, false, bfrag[k], (short)0, acc, false, false);
        }

        // Epilogue in C-matrix layout; 16 lanes share each edge index -> the
        // src/dst loads broadcast, and the Dh/Eh/Bh row gathers hit L2.
#pragma unroll
        for (int r = 0; r < 8; ++r) {
            const int m    = r + 8 * half;          // row within tile
            const int eidx = t * 16 + m;            // edge id
            const int s  = src[eidx];
            const int d2 = dst[eidx];
            const float gij = acc[r]
                            + Dh[(size_t)s  * D + col]
                            + Eh[(size_t)d2 * D + col];
            const float sg  = 1.0f / (1.0f + __expf(-gij));   // v_exp_f32
            e_out[(size_t)eidx * D + col] = shc[m * D + col] + fmaxf(gij, 0.0f);
            atomicAdd(&num[(size_t)d2 * D + col], sg * Bh[(size_t)s * D + col]);
            atomicAdd(&den[(size_t)d2 * D + col], sg);
        }

        wait_async0();
        __syncthreads();
        buf ^= 1;
    }
}

// ---------------------------------------------------------------------------
// h_out = h + relu(Ah + num / (den + 1e-6)), float4-vectorized elementwise.
// ---------------------------------------------------------------------------
__global__ __launch_bounds__(256)
void node_update_kernel(const float* __restrict__ h,
                        const float* __restrict__ Ah,
                        const float* __restrict__ num,
                        const float* __restrict__ den,
                        float* __restrict__ h_out,
                        int total4)
{
    const int i = blockIdx.x * blockDim.x + threadIdx.x;
    if (i >= total4) return;
    const float4 hv = ((const float4*)h)[i];
    const float4 av = ((const float4*)Ah)[i];
    const float4 nv = ((const float4*)num)[i];
    const float4 dv = ((const float4*)den)[i];
    float4 o;
    o.x = hv.x + fmaxf(av.x + nv.x / (dv.x + 1e-6f), 0.0f);
    o.y = hv.y + fmaxf(av.y + nv.y / (dv.y + 1e-6f), 0.0f);
    o.z = hv.z + fmaxf(av.z + nv.z / (dv.z + 1e-6f), 0.0f);
    o.w = hv.w + fmaxf(av.w + nv.w / (dv.w + 1e-6f), 0.0f);
    ((float4*)h_out)[i] = o;
}

// ---------------------------------------------------------------------------
// Host-side launcher. Inputs (setup_inputs order):
//  0:h 1:e 2:src 3:dst 4:WA 5:bA 6:WB 7:bB 8:WC 9:bC 10:WD 11:bD 12:WE 13:bE
// d_out = [h_out (N*D) | e_out (E*D)].
// Workspace: Ah | Bh | Dh | Eh | num | den   (6 * N*D floats = 153.6 MB).
// ---------------------------------------------------------------------------
extern "C" void kernel_launch(void* const* d_in, const int* in_sizes, int n_in,
                              void* d_out, int out_size, void* d_ws, size_t ws_size,
                              hipStream_t stream)
{
    const float* h   = (const float*)d_in[0];
    const float* e   = (const float*)d_in[1];
    const int*   src = (const int*)  d_in[2];
    const int*   dst = (const int*)  d_in[3];
    const float* WA  = (const float*)d_in[4];
    const float* bA  = (const float*)d_in[5];
    const float* WB  = (const float*)d_in[6];
    const float* bB  = (const float*)d_in[7];
    const float* WC  = (const float*)d_in[8];
    const float* bC  = (const float*)d_in[9];
    const float* WD  = (const float*)d_in[10];
    const float* bD  = (const float*)d_in[11];
    const float* WE  = (const float*)d_in[12];
    const float* bE  = (const float*)d_in[13];

    const int N = in_sizes[0] / D;
    const int E = in_sizes[2];
    const size_t ND = (size_t)N * D;

    float* ws   = (float*)d_ws;
    float* Ah   = ws;
    float* Bh   = ws + 1 * ND;
    float* Dh   = ws + 2 * ND;
    float* Eh   = ws + 3 * ND;
    float* numw = ws + 4 * ND;
    float* denw = ws + 5 * ND;

    float* h_out = (float*)d_out;
    float* e_out = (float*)d_out + ND;

    // Zero the scatter-sum accumulators (graph-capture safe).
    hipMemsetAsync(numw, 0, 2 * ND * sizeof(float), stream);

    const dim3 blk(256);

    // Four node projections; weight slices stay VGPR-resident per block.
    // 1024 blocks each: ~3 row tiles per block, machine fully occupied.
    node_proj_kernel<<<1024, blk, 0, stream>>>(h, WA, bA, Ah, N);
    node_proj_kernel<<<1024, blk, 0, stream>>>(h, WB, bB, Bh, N);
    node_proj_kernel<<<1024, blk, 0, stream>>>(h, WD, bD, Dh, N);
    node_proj_kernel<<<1024, blk, 0, stream>>>(h, WE, bE, Eh, N);

    // Fused edge GEMM + gate + residual + scatter: 2048 blocks to keep
    // enough waves in flight to hide L2 gather/atomic latency.
    edge_kernel<<<2048, blk, 0, stream>>>(e, src, dst, WC, bC,
                                          Bh, Dh, Eh, numw, denw, e_out, E);

    // Final node update.
    const int total4 = (int)(ND / 4);
    node_update_kernel<<<(total4 + 255) / 256, blk, 0, stream>>>(
        h, Ah, numw, denw, h_out, total4);
}